// TransformerBlockClassical_65481071408528
// MI455X (gfx1250) — compile-verified
//
#include <hip/hip_runtime.h>
#include <math.h>

#define E     8
#define H     4
#define FFN   2048
#define BATCH 8
#define SEQ   2048
#define NROWS (BATCH * SEQ)   // 16384
#define EPSLN 1e-5f

typedef __attribute__((ext_vector_type(2))) float v2f;
typedef __attribute__((ext_vector_type(8))) float v8f;

// ---------------------------------------------------------------------------
// Kernel 0: build zero-padded W2 (16 x FFN, rows 8..15 zero) so the FFN
// kernel's B-fragment loads are branch-free (no EXEC divergence in hot loop).
// ---------------------------------------------------------------------------
__global__ void pad_w2_kernel(const float* __restrict__ W2,
                              float* __restrict__ W2pad) {
    int i = blockIdx.x * blockDim.x + threadIdx.x;   // 0 .. 16*FFN-1
    int row = i / FFN;
    W2pad[i] = (row < E) ? W2[(size_t)row * FFN + (i % FFN)] : 0.f;
}

// ---------------------------------------------------------------------------
// Kernel 1: proj = x @ Wp^T   (8x8 matvec per row; Wp cached in LDS)
// ---------------------------------------------------------------------------
__global__ void proj_kernel(const float* __restrict__ x,
                            const float* __restrict__ Wp,
                            float* __restrict__ proj) {
    __shared__ float w[E * E];
    int tid = threadIdx.x;
    if (tid < E * E) w[tid] = Wp[tid];
    __syncthreads();
    int row = blockIdx.x * blockDim.x + tid;
    float xi[E];
#pragma unroll
    for (int k = 0; k < E; ++k) xi[k] = x[(size_t)row * E + k];
#pragma unroll
    for (int e = 0; e < E; ++e) {
        float s = 0.f;
#pragma unroll
        for (int k = 0; k < E; ++k) s += xi[k] * w[e * E + k];
        proj[(size_t)row * E + e] = s;
    }
}

// ---------------------------------------------------------------------------
// Kernel 2: fused flash-style self-attention, q=k=v=proj head slice (dk=2).
// One block per (b,h, 256-query chunk). K head slice (2048x2 f32 = 16 KB)
// staged in LDS; scores never materialized. Two-pass softmax => one exp per
// score element (134M v_exp_f32 total — the true bottleneck of this op).
// ---------------------------------------------------------------------------
__global__ void attn_kernel(const float* __restrict__ proj,
                            float* __restrict__ merged) {
    __shared__ float k0[SEQ];
    __shared__ float k1[SEQ];
    int b = blockIdx.x / H;
    int h = blockIdx.x % H;
    const float* base = proj + (size_t)b * SEQ * E + 2 * h;
    for (int i = threadIdx.x; i < SEQ; i += blockDim.x) {
        k0[i] = base[(size_t)i * E];
        k1[i] = base[(size_t)i * E + 1];
    }
    __syncthreads();
    int q = blockIdx.y * blockDim.x + threadIdx.x;
    float q0 = k0[q], q1 = k1[q];
    const float scale = 0.70710678118654752f;   // 1/sqrt(dk), dk=2
    float mx = -3.402823466e38f;
    for (int k = 0; k < SEQ; ++k) {
        float s = (q0 * k0[k] + q1 * k1[k]) * scale;
        mx = fmaxf(mx, s);
    }
    float sum = 0.f, a0 = 0.f, a1 = 0.f;
    for (int k = 0; k < SEQ; ++k) {
        float s = (q0 * k0[k] + q1 * k1[k]) * scale;
        float p = __expf(s - mx);
        sum += p;
        a0 += p * k0[k];
        a1 += p * k1[k];
    }
    float inv = 1.f / sum;
    merged[((size_t)b * SEQ + q) * E + 2 * h]     = a0 * inv;
    merged[((size_t)b * SEQ + q) * E + 2 * h + 1] = a1 * inv;
}

// ---------------------------------------------------------------------------
// Kernel 3: attn_out = merged @ Wo^T ; x1 = LN(x + attn_out) ;
//           out_q = cumprod(cos(x1))  — all per-row, fully fused.
// ---------------------------------------------------------------------------
__global__ void postattn_kernel(const float* __restrict__ x,
                                const float* __restrict__ merged,
                                const float* __restrict__ Wo,
                                const float* __restrict__ g1,
                                const float* __restrict__ b1,
                                float* __restrict__ x1,
                                float* __restrict__ outq) {
    __shared__ float w[E * E];
    __shared__ float sg[E], sb[E];
    int tid = threadIdx.x;
    if (tid < E * E) w[tid] = Wo[tid];
    if (tid < E) { sg[tid] = g1[tid]; sb[tid] = b1[tid]; }
    __syncthreads();
    int row = blockIdx.x * blockDim.x + tid;
    float m[E], y[E];
#pragma unroll
    for (int k = 0; k < E; ++k) m[k] = merged[(size_t)row * E + k];
    float mu = 0.f;
#pragma unroll
    for (int e = 0; e < E; ++e) {
        float s = x[(size_t)row * E + e];
#pragma unroll
        for (int k = 0; k < E; ++k) s += m[k] * w[e * E + k];
        y[e] = s;
        mu += s;
    }
    mu *= (1.f / E);
    float var = 0.f;
#pragma unroll
    for (int e = 0; e < E; ++e) { float d = y[e] - mu; var += d * d; }
    var *= (1.f / E);
    float rinv = rsqrtf(var + EPSLN);
    float cp = 1.f;
#pragma unroll
    for (int e = 0; e < E; ++e) {
        float v = (y[e] - mu) * rinv * sg[e] + sb[e];
        x1[(size_t)row * E + e] = v;
        cp *= __cosf(v);
        outq[(size_t)row * E + e] = cp;
    }
}

// ---------------------------------------------------------------------------
// Kernel 4: fused FFN on the WMMA path (V_WMMA_F32_16X16X4_F32).
//   h   = relu(out_q @ W1^T + bb1)       (16384x8)@(8x2048)   K=8  -> 2 WMMA
//   ffn = h @ W2pad^T + bb2              (16384x2048)@(2048x16) K=2048
//   out = LN(x1 + ffn)
// One wave per 16-row M-tile. The 16xFFN h-tile never touches HBM: each
// 16-col chunk is produced in D-layout, ReLU'd, bounced through LDS into
// A-layout, and immediately consumed by the second GEMM's accumulator.
// W2pad rows 8..15 are zero, so all 32 lanes load unconditionally —
// the inner loop is straight-line (no saveexec/branches around WMMA).
// 32-bit A 16x4 layout: lanes 0-15 hold {K0,K1}, lanes 16-31 hold {K2,K3};
// B assumed symmetric (N across lanes, K split per lane-half).
// ---------------------------------------------------------------------------
__global__ void __launch_bounds__(32) ffn_wmma_kernel(
    const float* __restrict__ x1, const float* __restrict__ outq,
    const float* __restrict__ W1, const float* __restrict__ bb1,
    const float* __restrict__ W2pad, const float* __restrict__ bb2,
    const float* __restrict__ g2, const float* __restrict__ b2,
    float* __restrict__ out) {
    __shared__ float hTile[16 * 16];   // h chunk, row-major
    __shared__ float fTile[16 * E];    // ffn pre-norm result

    int lane = threadIdx.x;
    int half = lane >> 4;      // 0: lanes 0-15, 1: lanes 16-31
    int mn   = lane & 15;      // M for A-frags, N for B-frags
    int row0 = blockIdx.x * 16;

    // A fragments for GEMM1 (out_q tile, K=8 -> two K=4 steps)
    v2f a0, a1;
    {
        const float* ap = outq + (size_t)(row0 + mn) * E + 2 * half;
        a0.x = ap[0]; a0.y = ap[1];   // K-step 0: K = 2*half, 2*half+1
        a1.x = ap[4]; a1.y = ap[5];   // K-step 1: K = 4+2*half, ...
    }

    const float* w2row = W2pad + (size_t)mn * FFN + 2 * half;  // branch-free B
    v8f cacc = {};   // 16x16 output accumulator (cols 8..15 are zero pad)

    for (int c = 0; c < FFN / 16; ++c) {
        int col = c * 16 + mn;        // FFN column handled by this lane (B/N)

        // ---- GEMM1: hD = out_q(16x8) @ W1chunk^T(8x16) ----
        v2f wb0, wb1;
        {
            const float* wp = W1 + (size_t)col * E + 2 * half;
            wb0.x = wp[0]; wb0.y = wp[1];
            wb1.x = wp[4]; wb1.y = wp[5];
            __builtin_prefetch(wp + 16 * E, 0, 0);   // next chunk of W1
        }
        v8f hD = {};
        hD = __builtin_amdgcn_wmma_f32_16x16x4_f32(false, a0, false, wb0,
                                                   (short)0, hD, false, false);
        hD = __builtin_amdgcn_wmma_f32_16x16x4_f32(false, a1, false, wb1,
                                                   (short)0, hD, false, false);

        // bias + ReLU in D-layout, spill chunk to LDS (D-layout -> row-major)
        float bias = bb1[col];
#pragma unroll
        for (int r = 0; r < 8; ++r) {
            float hv = hD[r] + bias;                    // elem (r+8*half, mn)
            hTile[(r + 8 * half) * 16 + mn] = fmaxf(hv, 0.f);
        }
        __syncthreads();

        // ---- GEMM2: cacc += h(16x16) @ W2chunk^T(16x16-pad), 4 K-steps ----
#pragma unroll
        for (int kk = 0; kk < 4; ++kk) {
            v2f a2, b2v;
            const float* hp = &hTile[mn * 16 + 4 * kk + 2 * half];
            a2.x = hp[0]; a2.y = hp[1];
            const float* w2p = w2row + c * 16 + 4 * kk;
            b2v.x = w2p[0]; b2v.y = w2p[1];
            cacc = __builtin_amdgcn_wmma_f32_16x16x4_f32(
                false, a2, false, b2v, (short)0, cacc, false, false);
        }
        __syncthreads();   // before hTile is overwritten next iteration
    }

    // cacc element (m = r + 8*half, e = mn), valid for e < 8
    if (mn < E) {
        float be = bb2[mn];
#pragma unroll
        for (int r = 0; r < 8; ++r)
            fTile[(r + 8 * half) * E + mn] = cacc[r] + be;
    }
    __syncthreads();

    // Final residual + LayerNorm: lanes 0-15 each own one row of 8.
    if (lane < 16) {
        int m = lane;
        float y[E];
        float mu = 0.f;
#pragma unroll
        for (int e = 0; e < E; ++e) {
            y[e] = x1[(size_t)(row0 + m) * E + e] + fTile[m * E + e];
            mu += y[e];
        }
        mu *= (1.f / E);
        float var = 0.f;
#pragma unroll
        for (int e = 0; e < E; ++e) { float d = y[e] - mu; var += d * d; }
        var *= (1.f / E);
        float rinv = rsqrtf(var + EPSLN);
#pragma unroll
        for (int e = 0; e < E; ++e)
            out[(size_t)(row0 + m) * E + e] =
                (y[e] - mu) * rinv * g2[e] + b2[e];
    }
}

// ---------------------------------------------------------------------------
extern "C" void kernel_launch(void* const* d_in, const int* in_sizes, int n_in,
                              void* d_out, int out_size, void* d_ws,
                              size_t ws_size, hipStream_t stream) {
    const float* x   = (const float*)d_in[0];
    const float* Wp  = (const float*)d_in[1];
    const float* Wo  = (const float*)d_in[2];
    const float* g1  = (const float*)d_in[3];
    const float* b1  = (const float*)d_in[4];
    const float* W1  = (const float*)d_in[5];
    const float* bb1 = (const float*)d_in[6];
    const float* W2  = (const float*)d_in[7];
    const float* bb2 = (const float*)d_in[8];
    const float* g2  = (const float*)d_in[9];
    const float* b2  = (const float*)d_in[10];
    float* out = (float*)d_out;

    float* ws     = (float*)d_ws;
    float* proj   = ws;                          // N*E
    float* merged = ws + (size_t)NROWS * E;      // N*E
    float* x1v    = ws + (size_t)2 * NROWS * E;  // N*E
    float* outqv  = ws + (size_t)3 * NROWS * E;  // N*E
    float* w2pad  = ws + (size_t)4 * NROWS * E;  // 16*FFN  (~2.13 MB total)

    pad_w2_kernel<<<(16 * FFN) / 256, 256, 0, stream>>>(W2, w2pad);
    proj_kernel<<<NROWS / 256, 256, 0, stream>>>(x, Wp, proj);
    attn_kernel<<<dim3(BATCH * H, SEQ / 256), 256, 0, stream>>>(proj, merged);
    postattn_kernel<<<NROWS / 256, 256, 0, stream>>>(x, merged, Wo, g1, b1,
                                                     x1v, outqv);
    ffn_wmma_kernel<<<NROWS / 16, 32, 0, stream>>>(x1v, outqv, W1, bb1, w2pad,
                                                   bb2, g2, b2, out);
}